// OEQConvolution_9363028705518
// MI455X (gfx1250) — compile-verified
//
#include <hip/hip_runtime.h>
#include <hip/hip_bf16.h>
#include <stdint.h>

// Problem constants (from the reference).
#define N_EDGES   1600000
#define MUL       32
#define WPE       160           // weight floats per edge = N_PATHS * MUL
#define ROW       128           // floats per node row / per output row
#define INV_SQRT2 0.70710678118654752440f
#define INV_SQRT3 0.57735026918962576451f

#define WAVES_PER_BLOCK 8
#define THREADS   (WAVES_PER_BLOCK * 32)
#define NBLOCKS   2048

typedef __attribute__((ext_vector_type(4))) float v4f;  // native vector for NT load

__global__ void oeq_zero_kernel(float* __restrict__ out, int n) {
  int i = blockIdx.x * blockDim.x + threadIdx.x;
  int stride = gridDim.x * blockDim.x;
  for (; i < n; i += stride) out[i] = 0.0f;
}

// Issue the async DMA of one edge's 640B weight tile into an LDS buffer.
// Streamed once -> non-temporal so it never evicts the L2-resident x/out rows.
#define ASYNC_WTILE(ldst_a, ldst_b, gbase)                                    \
  do {                                                                        \
    const uint64_t _g0 = (gbase) + (uint64_t)lane * 16u;                      \
    const uint64_t _g1 = (gbase) + 512u + (uint64_t)lane * 4u;                \
    asm volatile("global_load_async_to_lds_b128 %0, %1, off th:TH_LOAD_NT"    \
                 :: "v"(ldst_a), "v"(_g0) : "memory");                        \
    asm volatile("global_load_async_to_lds_b32 %0, %1, off th:TH_LOAD_NT"     \
                 :: "v"(ldst_b), "v"(_g1) : "memory");                        \
  } while (0)

// One wave (32 lanes) per edge; lane == channel u. Double-buffered async
// weight staging: tile for edge e+stride is in flight while edge e computes.
__global__ __launch_bounds__(THREADS) void oeq_conv_kernel(
    const float* __restrict__ x,            // [N, 128]
    const float* __restrict__ edge_filter,  // [E, 4]
    const float* __restrict__ weight,       // [E, 160]
    const int*   __restrict__ edge_src,     // [E]
    const int*   __restrict__ edge_dst,     // [E]
    float*       __restrict__ out)          // [N, 128], pre-zeroed
{
  __shared__ float smem[WAVES_PER_BLOCK * 2 * WPE];

  const int lane  = threadIdx.x & 31;
  const int wave  = threadIdx.x >> 5;
  const int gwave = blockIdx.x * WAVES_PER_BLOCK + wave;
  const int nwaves = gridDim.x * WAVES_PER_BLOCK;

  // Byte addresses (wave-relative LDS offsets) for the async DMA destinations.
  // Generic->LDS: low 32 bits of a shared-aperture address are the LDS offset.
  const unsigned lds_base = (unsigned)(uintptr_t)(&smem[wave * 2 * WPE]);
  unsigned ldsa[2], ldsb[2];
#pragma unroll
  for (int b = 0; b < 2; ++b) {
    ldsa[b] = lds_base + (unsigned)b * (WPE * 4u) + (unsigned)lane * 16u;         // b128 dest
    ldsb[b] = lds_base + (unsigned)b * (WPE * 4u) + 512u + (unsigned)lane * 4u;   // b32 dest
  }
  const int wslot = wave * 2 * WPE;  // float index of this wave's buffer pair

  int e = gwave;
  int cur = 0;
  if (e < N_EDGES) {
    ASYNC_WTILE(ldsa[0], ldsb[0], (uint64_t)(uintptr_t)(weight + (size_t)e * WPE));
  }

  for (; e < N_EDGES; e += nwaves) {
    // ---- keep the pipeline full: launch next edge's tile into other buffer ----
    const int en = e + nwaves;
    const bool has_next = (en < N_EDGES);
    if (has_next) {
      ASYNC_WTILE(ldsa[cur ^ 1], ldsb[cur ^ 1],
                  (uint64_t)(uintptr_t)(weight + (size_t)en * WPE));
    }

    // ---- while DMAs fly: gather node features + edge scalars ----
    const int eu  = __builtin_amdgcn_readfirstlane(e);
    const int src = __builtin_nontemporal_load(edge_src + eu);
    const int dst = __builtin_nontemporal_load(edge_dst + eu);

    const float* xrow = x + (size_t)src * ROW;   // L2-resident working set
    const float xs0  = xrow[lane];
    const float xs1x = xrow[MUL + 3 * lane + 0];
    const float xs1y = xrow[MUL + 3 * lane + 1];
    const float xs1z = xrow[MUL + 3 * lane + 2];

    const v4f efv = __builtin_nontemporal_load(
        reinterpret_cast<const v4f*>(edge_filter) + eu);
    const float y0  = efv.x;
    const float y1x = efv.y;
    const float y1y = efv.z;
    const float y1z = efv.w;

    // Prefetch next edge's gathered x row into near caches (global_prefetch_b8).
    if (has_next) {
      const int nsrc = edge_src[__builtin_amdgcn_readfirstlane(en)];
      __builtin_prefetch((const char*)(x + (size_t)nsrc * ROW) + lane * 16, 0, 3);
    }

    // ---- wait only for the *current* buffer's 2 async ops (in-order done) ----
    if (has_next) {
      asm volatile("s_wait_asynccnt 0x2" ::: "memory");
    } else {
      asm volatile("s_wait_asynccnt 0x0" ::: "memory");
    }
    // Read weights straight off the __shared__ array so codegen stays on the
    // ds_load path (indexing through generic pointers demotes these to FLAT).
    const int wb = wslot + cur * WPE;
    const float w00   = smem[wb + 0 * MUL + lane];
    const float w11_0 = smem[wb + 1 * MUL + lane];
    const float w01   = smem[wb + 2 * MUL + lane];
    const float w10   = smem[wb + 3 * MUL + lane];
    const float w11_1 = smem[wb + 4 * MUL + lane];

    // ---- equivariant products ----
    const float dot = xs1x * y1x + xs1y * y1y + xs1z * y1z;
    const float o0  = INV_SQRT2 * (w00 * xs0 * y0 + w11_0 * (INV_SQRT3 * dot));

    const float cx = xs1y * y1z - xs1z * y1y;
    const float cy = xs1z * y1x - xs1x * y1z;
    const float cz = xs1x * y1y - xs1y * y1x;

    const float a = INV_SQRT2;               // outer scale
    const float b = INV_SQRT2 * INV_SQRT2;   // outer * inner cross scale = 0.5
    const float o1x = a * (w01 * xs0 * y1x + w10 * xs1x * y0) + b * (w11_1 * cx);
    const float o1y = a * (w01 * xs0 * y1y + w10 * xs1y * y0) + b * (w11_1 * cy);
    const float o1z = a * (w01 * xs0 * y1z + w10 * xs1z * y0) + b * (w11_1 * cz);

    // ---- scatter-add into the L2-resident output ----
    float* orow = out + (size_t)dst * ROW;
    atomicAdd(orow + lane, o0);
    atomicAdd(orow + MUL + 3 * lane + 0, o1x);
    atomicAdd(orow + MUL + 3 * lane + 1, o1y);
    atomicAdd(orow + MUL + 3 * lane + 2, o1z);

    cur ^= 1;
  }
}

extern "C" void kernel_launch(void* const* d_in, const int* in_sizes, int n_in,
                              void* d_out, int out_size, void* d_ws, size_t ws_size,
                              hipStream_t stream) {
  const float* x    = (const float*)d_in[0];
  const float* ef   = (const float*)d_in[1];
  const float* w    = (const float*)d_in[2];
  const int*   esrc = (const int*)d_in[3];
  const int*   edst = (const int*)d_in[4];
  float* out = (float*)d_out;

  // Output is accumulated via atomics: zero it first (harness poisons it).
  oeq_zero_kernel<<<512, 256, 0, stream>>>(out, out_size);
  oeq_conv_kernel<<<NBLOCKS, THREADS, 0, stream>>>(x, ef, w, esrc, edst, out);
}